// Yolo_v3_85478439125832
// MI455X (gfx1250) — compile-verified
//
#include <hip/hip_runtime.h>
#include <hip/hip_bf16.h>

#define NUM_CLASSES 20
#define HWA_N 22743          // (5776+1444+361)*3
#define G_PER 8
#define MAXREC 80            // 8 gt * (9 ignores + 1 target)
#define RECF 12
#define ACC_N 16             // float offset where records begin in ws
// acc layout (floats): [0]=S(neg baseline) [1]=A(conf_pos) [2]=R(removed)
//                      [3]=C(cls) [4]=D(txty) [5]=E(twth) [6]=npos

typedef __attribute__((ext_vector_type(2))) float v2f;
typedef __attribute__((ext_vector_type(8))) float v8f;

__device__ __forceinline__ float sigm_(float x) { return 1.0f / (1.0f + expf(-x)); }
__device__ __forceinline__ float softplus_(float x) {
    // logaddexp(0, x) = max(x,0) + log1p(exp(-|x|))
    return fmaxf(x, 0.0f) + log1pf(expf(-fabsf(x)));
}

// ---------------- init: zero accumulators ----------------
__global__ void yolo_init_k(float* ws) {
    if (threadIdx.x < 8) ws[threadIdx.x] = 0.0f;
}

// ---------------- baseline: S = sum sigmoid(pconf)^2 ----------------
__global__ void yolo_negsum_k(const float* __restrict__ pconf, float* ws, int n) {
    int tid = blockIdx.x * blockDim.x + threadIdx.x;
    int stride = gridDim.x * blockDim.x;
    int n4 = n >> 2;
    const float4* p4 = (const float4*)pconf;
    float acc = 0.0f;
    for (int i = tid; i < n4; i += stride) {
        float4 v = p4[i];
        float s0 = sigm_(v.x), s1 = sigm_(v.y), s2 = sigm_(v.z), s3 = sigm_(v.w);
        acc += s0 * s0 + s1 * s1 + s2 * s2 + s3 * s3;
    }
    int rem = n - (n4 << 2);
    if (tid < rem) {
        float s = sigm_(pconf[(n4 << 2) + tid]);
        acc += s * s;
    }
    // ---- wave32 reduction ----
#if defined(__gfx1250__) && __has_builtin(__builtin_amdgcn_wmma_f32_16x16x4_f32)
    // Layout-robust WMMA sum: D = A(16x4) * ones(4x16). Each lane's {acc,0}
    // occupies two distinct A elements; every column of D therefore equals
    // sum over ALL 32 lane partials. Column n is split across lanes n / n+16
    // (VGPRs 0..7 each), so per-lane Σd + shfl_xor(16) = wave total.
    v2f a; a[0] = acc; a[1] = 0.0f;
    v2f b; b[0] = 1.0f; b[1] = 1.0f;
    v8f c = {};
    v8f d = __builtin_amdgcn_wmma_f32_16x16x4_f32(false, a, false, b, (short)0, c,
                                                  false, false);
    float s = d[0] + d[1] + d[2] + d[3] + d[4] + d[5] + d[6] + d[7];
    s += __shfl_xor(s, 16, 32);
#else
    float s = acc;
    for (int off = 16; off; off >>= 1) s += __shfl_xor(s, off, 32);
#endif
    if ((threadIdx.x & 31) == 0) atomicAdd(ws + 0, s);
}

// ---------------- matching: one thread per image ----------------
__global__ void yolo_match_k(const float* __restrict__ ptxywh,
                             const float* __restrict__ gboxes,
                             const int* __restrict__ glabels,
                             float* ws, int B) {
    int img = blockIdx.x * blockDim.x + threadIdx.x;
    if (img >= B) return;

    const float ancw[9] = {0.02f, 0.04f, 0.08f, 0.15f, 0.2f, 0.35f, 0.45f, 0.6f, 0.85f};
    const float anch[9] = {0.03f, 0.06f, 0.12f, 0.2f, 0.35f, 0.25f, 0.55f, 0.7f, 0.9f};
    const float grid_a[9] = {76.f, 76.f, 76.f, 38.f, 38.f, 38.f, 19.f, 19.f, 19.f};
    const float off_a[9] = {0.f, 0.f, 0.f, 5776.f, 5776.f, 5776.f, 7220.f, 7220.f, 7220.f};

    int widx[MAXREC];
    unsigned char wtype[MAXREC]; // 1 = ignore, 2 = target
    signed char wslot[MAXREC];   // gt index for targets
    float tpay[G_PER][7];        // conf, lab, tx, ty, tw, th, weight
    int nw = 0;

    for (int i = 0; i < G_PER; ++i) {
        const float* gb = gboxes + ((long long)img * G_PER + i) * 4;
        float l = gb[0], t = gb[1], r = gb[2], bm = gb[3];
        float cx = (l + r) * 0.5f, cy = (t + bm) * 0.5f;
        float w = r - l, h = bm - t;
        float area_g = w * h;

        int best = 0;
        float bestiou = -1.0f;
        int md_a[9];
        float crx_a[9], cry_a[9];
        for (int a = 0; a < 9; ++a) {
            float aw = ancw[a], ah = anch[a];
            // IoU of zero-centered boxes (exactly as reference arithmetic)
            float ltx = fmaxf(-0.5f * w, -0.5f * aw);
            float lty = fmaxf(-0.5f * h, -0.5f * ah);
            float rbx = fminf(0.5f * w, 0.5f * aw);
            float rby = fminf(0.5f * h, 0.5f * ah);
            float iw = fmaxf(rbx - ltx, 0.0f), ih = fmaxf(rby - lty, 0.0f);
            float inter = iw * ih;
            float iou = inter / (area_g + aw * ah - inter + 1e-16f);

            float g = grid_a[a];
            int col = (int)(cx * g);
            int row = (int)(cy * g);
            int md = (int)((off_a[a] + (float)row * g + (float)col) * 3.0f);
            md_a[a] = md;
            crx_a[a] = (float)col;
            cry_a[a] = (float)row;
            if (iou > 0.5f) { widx[nw] = md; wtype[nw] = 1; wslot[nw] = -1; ++nw; }
            if (iou > bestiou) { bestiou = iou; best = a; } // first-max like jnp.argmax
        }
        // target write for best anchor
        float g = grid_a[best];
        float crx = crx_a[best], cry = cry_a[best];
        float tx = (cx - crx / g) * g;
        float ty = (cy - cry / g) * g;
        float aw = ancw[best], ah = anch[best];
        float tw = logf(r / aw);   // NOTE: reference uses ltrb[2:], i.e. (r, b)
        float th = logf(bm / ah);
        // NOTE: reference indexes pt_b[best] -> row = anchor index (0..8)
        const float* pt = ptxywh + ((long long)img * HWA_N + best) * 4;
        float px = sigm_(pt[0]), py = sigm_(pt[1]);
        float pxc = (px + crx) / g, pyc = (py + cry) / g;
        float pw = expf(pt[2]) * aw, ph = expf(pt[3]) * ah;
        float pl = pxc - pw * 0.5f, ptp = pyc - ph * 0.5f;
        float pr = pxc + pw * 0.5f, pb = pyc + ph * 0.5f;
        float ltx = fmaxf(pl, l), lty = fmaxf(ptp, t);
        float rbx = fminf(pr, r), rby = fminf(pb, bm);
        float iw = fmaxf(rbx - ltx, 0.0f), ih = fmaxf(rby - lty, 0.0f);
        float inter = iw * ih;
        float area_p = (pr - pl) * (pb - ptp);
        float area_b = (r - l) * (bm - t);
        float conf = inter / (area_p + area_b - inter + 1e-16f);
        float weight = 2.0f - r * bm;
        int lab = glabels[(long long)img * G_PER + i] - 1;

        tpay[i][0] = conf; tpay[i][1] = (float)lab;
        tpay[i][2] = tx; tpay[i][3] = ty; tpay[i][4] = tw; tpay[i][5] = th;
        tpay[i][6] = weight;
        widx[nw] = md_a[best]; wtype[nw] = 2; wslot[nw] = (signed char)i; ++nw;
    }

    // last-write-wins per index -> final records
    float* rec = ws + ACC_N + (long long)img * MAXREC * RECF;
    int nrec = 0;
    float nposf = 0.0f;
    for (int k = 0; k < nw; ++k) {
        bool last = true;
        for (int k2 = k + 1; k2 < nw; ++k2)
            if (widx[k2] == widx[k]) { last = false; break; }
        if (!last) continue;
        float* s = rec + nrec * RECF;
        s[0] = (float)wtype[k];
        s[1] = (float)widx[k];
        if (wtype[k] == 2) {
            int i = wslot[k];
            for (int q = 0; q < 7; ++q) s[2 + q] = tpay[i][q];
            nposf += 1.0f;
        } else {
            for (int q = 0; q < 7; ++q) s[2 + q] = 0.0f;
        }
        ++nrec;
    }
    for (int k = nrec; k < MAXREC; ++k) rec[k * RECF] = 0.0f; // invalidate rest
    if (nposf > 0.0f) atomicAdd(ws + 6, nposf);
}

// ---------------- corrections: one thread per record slot ----------------
__global__ void yolo_correct_k(const float* __restrict__ pconf,
                               const float* __restrict__ pcls,
                               const float* __restrict__ ptxywh,
                               float* ws, int B) {
    int gidx = blockIdx.x * blockDim.x + threadIdx.x;
    if (gidx >= B * MAXREC) return;
    int img = gidx / MAXREC, slot = gidx % MAXREC;
    const float* s = ws + ACC_N + ((long long)img * MAXREC + slot) * RECF;
    float type = s[0];
    if (type < 0.5f) return;

    int idx = (int)s[1];
    long long base = (long long)img * HWA_N + idx;
    float sg = sigm_(pconf[base]);
    float removed = sg * sg; // this anchor leaves the "neg" set (mneg=0)

    float accA = 0.0f, accC = 0.0f, accD = 0.0f, accE = 0.0f;
    if (type > 1.5f) { // target: mpos=1
        float conf = s[2];
        int lab = (int)s[3];
        float d0 = sg - conf;
        accA = d0 * d0;
        const float* pcl = pcls + base * NUM_CLASSES;
        for (int c = 0; c < NUM_CLASSES; ++c) {
            float x = pcl[c];
            accC += softplus_(x) - x * ((c == lab) ? 1.0f : 0.0f);
        }
        const float* pt = ptxywh + base * 4;
        float tx = s[4], ty = s[5], tw = s[6], th = s[7], wgt = s[8];
        float b0 = softplus_(pt[0]) - pt[0] * tx;
        float b1 = softplus_(pt[1]) - pt[1] * ty;
        accD = (b0 + b1) * wgt;
        float dw = pt[2] - tw, dh = pt[3] - th;
        accE = (dw * dw + dh * dh) * wgt;
        atomicAdd(ws + 1, accA);
        atomicAdd(ws + 3, accC);
        atomicAdd(ws + 4, accD);
        atomicAdd(ws + 5, accE);
    }
    atomicAdd(ws + 2, removed);
}

// ---------------- finalize ----------------
__global__ void yolo_final_k(const float* __restrict__ ws, float* out, int B) {
    float S = ws[0], A = ws[1], R = ws[2], C = ws[3], D = ws[4], E = ws[5];
    float npos = fmaxf(ws[6], 1.0f);
    float Bf = (float)B;
    // l_conf_pos*1.0 + l_conf_neg*0.5 + l_cls + l_txty + l_twth
    out[0] = A / Bf + 0.5f * (S - R) / Bf + C / npos + D / Bf + E / Bf;
}

extern "C" void kernel_launch(void* const* d_in, const int* in_sizes, int n_in,
                              void* d_out, int out_size, void* d_ws, size_t ws_size,
                              hipStream_t stream) {
    const float* pconf = (const float*)d_in[0];
    const float* pcls = (const float*)d_in[1];
    const float* ptxywh = (const float*)d_in[2];
    const float* gboxes = (const float*)d_in[3];
    const int* glabels = (const int*)d_in[4];
    float* ws = (float*)d_ws;
    float* out = (float*)d_out;

    int B = in_sizes[0] / HWA_N;
    int n = B * HWA_N;

    yolo_init_k<<<1, 32, 0, stream>>>(ws);
    yolo_negsum_k<<<512, 256, 0, stream>>>(pconf, ws, n);
    yolo_match_k<<<(B + 63) / 64, 64, 0, stream>>>(ptxywh, gboxes, glabels, ws, B);
    yolo_correct_k<<<(B * MAXREC + 255) / 256, 256, 0, stream>>>(pconf, pcls, ptxywh, ws, B);
    yolo_final_k<<<1, 1, 0, stream>>>(ws, out, B);
    (void)n_in; (void)out_size; (void)ws_size;
}